// PseudoBlockSparseLinear_17849884082261
// MI455X (gfx1250) — compile-verified
//
#include <hip/hip_runtime.h>

typedef __attribute__((ext_vector_type(2))) float v2f;
typedef __attribute__((ext_vector_type(8))) float v8f;

#define K_DIM   4096   // D_IN
#define N_DIM   4096   // D_OUT
#define M_DIM   4096   // B*S = 2*2048
#define BLK     32     // mask block size
#define M_TILE  256    // rows per workgroup (32 per wave: 2 x 16-row WMMA subtiles)
#define N_TILE  32     // one mask-block column -> uniform skip across workgroup
#define LDSW    36     // padded LDS row stride (floats): 16B-aligned rows, conflict-free
#define NKB     (K_DIM / BLK)

// ---- CDNA5 async global->LDS path (ASYNCcnt), if the toolchain exposes it ----
#if defined(__has_builtin)
#if __has_builtin(__builtin_amdgcn_global_load_async_to_lds_b128) && \
    __has_builtin(__builtin_amdgcn_s_wait_asynccnt)
#define USE_ASYNC 1
#endif
#endif
#ifndef USE_ASYNC
#define USE_ASYNC 0
#endif

#if USE_ASYNC
// Builtin signature (from clang diagnostic): param0 = 16B int-vector pointer in AS(1),
// param1 = matching pointer in AS(3), then imm offset / imm cpol.
typedef int v4i __attribute__((vector_size(16)));
typedef v4i __attribute__((address_space(1)))* gvp_t;
typedef v4i __attribute__((address_space(3)))* lvp_t;
#define NBUF 2
#else
#define NBUF 1
#endif

__global__ __launch_bounds__(256)
void bsl_wmma_f32_kernel(const float* __restrict__ x,
                         const float* __restrict__ w,
                         const float* __restrict__ bias,
                         const int*  __restrict__ mask,
                         float* __restrict__ out)
{
    __shared__ float Alds[NBUF][M_TILE][LDSW]; // x tile [m][k]
    __shared__ float Blds[NBUF][BLK][LDSW];    // w tile [n][k]  (B[k][n] = w[n][k])

    const int nb   = blockIdx.x;           // N block (32 wide)
    const int mb   = blockIdx.y;           // M block (256 tall)
    const int t    = threadIdx.x;
    const int wave = t >> 5;               // 0..7
    const int lane = t & 31;

    const int m0 = mb * M_TILE;
    const int n0 = nb * N_TILE;

    // 2 M-subtiles x 2 N-subtiles per wave
    v8f acc00 = {}, acc01 = {}, acc10 = {}, acc11 = {};

    // global->LDS load mapping: 256 threads, float4 granularity
    const int lr = t >> 3;                 // 0..31 row within pass
    const int lk = (t & 7) << 2;           // 0,4,...,28

    // WMMA fragment mapping (wave32)
    const int mrow = wave * 32;            // wave's 32-row M strip
    const int am   = lane & 15;            // M (A) / N (B,C,D) within 16
    const int kg   = (lane >> 4) << 1;     // K pair base: 0 or 2

    const size_t maskrow = (size_t)n0 * K_DIM;

    // ---- per-buffer compute: 8 K-steps of V_WMMA_F32_16X16X4_F32, 2x2 subtiles ----
    auto compute = [&](int buf) {
        #pragma unroll
        for (int kk = 0; kk < 8; ++kk) {
            const int kc = kk * 4 + kg;
            v2f a0 = *(const v2f*)&Alds[buf][mrow + am][kc];
            v2f a1 = *(const v2f*)&Alds[buf][mrow + 16 + am][kc];
            v2f b0 = *(const v2f*)&Blds[buf][am][kc];
            v2f b1 = *(const v2f*)&Blds[buf][am + 16][kc];
            acc00 = __builtin_amdgcn_wmma_f32_16x16x4_f32(false, a0, false, b0, (short)0, acc00, false, false);
            acc01 = __builtin_amdgcn_wmma_f32_16x16x4_f32(false, a0, false, b1, (short)0, acc01, false, false);
            acc10 = __builtin_amdgcn_wmma_f32_16x16x4_f32(false, a1, false, b0, (short)0, acc10, false, false);
            acc11 = __builtin_amdgcn_wmma_f32_16x16x4_f32(false, a1, false, b1, (short)0, acc11, false, false);
        }
    };

#if USE_ASYNC
    // 9 async b128 copies per thread per K-block (1 B-tile + 8 A-tile passes)
    auto issue = [&](int kb, int buf) {
        const int k0 = kb * BLK;
        __builtin_amdgcn_global_load_async_to_lds_b128(
            (gvp_t)(w + (size_t)(n0 + lr) * K_DIM + k0 + lk),
            (lvp_t)&Blds[buf][lr][lk], 0, 0);
        #pragma unroll
        for (int p = 0; p < 8; ++p) {
            const int r = lr + p * 32;
            __builtin_amdgcn_global_load_async_to_lds_b128(
                (gvp_t)(x + (size_t)(m0 + r) * K_DIM + k0 + lk),
                (lvp_t)&Alds[buf][r][lk], 0, 0);
        }
    };

    // double-buffered pipeline over active K-blocks (uniform mask scan -> scalar)
    int kb = 0;
    while (kb < NKB && mask[maskrow + (size_t)kb * BLK] == 0) ++kb;
    int buf = 0;
    if (kb < NKB) issue(kb, buf);
    while (kb < NKB) {
        int kbn = kb + 1;
        while (kbn < NKB && mask[maskrow + (size_t)kbn * BLK] == 0) ++kbn;
        if (kbn < NKB) {
            issue(kbn, buf ^ 1);                    // prefetch next active block
            __builtin_amdgcn_s_wait_asynccnt(9);    // current 9 done (in-order), next 9 in flight
        } else {
            __builtin_amdgcn_s_wait_asynccnt(0);
        }
        __syncthreads();        // all waves' async writes visible
        compute(buf);
        __syncthreads();        // all waves done reading before buffer reuse
        buf ^= 1;
        kb = kbn;
    }
#else
    // fallback: synchronous global->VGPR->LDS staging, single buffer
    for (int kb = 0; kb < NKB; ++kb) {
        if (mask[maskrow + (size_t)kb * BLK] == 0) continue;
        const int k0 = kb * BLK;

        __syncthreads();
        {
            const float4 wv = *(const float4*)(w + (size_t)(n0 + lr) * K_DIM + k0 + lk);
            *(v2f*)&Blds[0][lr][lk + 0] = (v2f){wv.x, wv.y};
            *(v2f*)&Blds[0][lr][lk + 2] = (v2f){wv.z, wv.w};
        }
        #pragma unroll
        for (int p = 0; p < 8; ++p) {
            const int r = lr + p * 32;
            const float4 xv = *(const float4*)(x + (size_t)(m0 + r) * K_DIM + k0 + lk);
            *(v2f*)&Alds[0][r][lk + 0] = (v2f){xv.x, xv.y};
            *(v2f*)&Alds[0][r][lk + 2] = (v2f){xv.z, xv.w};
        }
        __syncthreads();
        compute(0);
    }
#endif

    // ---- epilogue: bias add + store ----
    // C/D layout: VGPR r -> M = r (lanes 0-15) or r+8 (lanes 16-31); N = lane%16.
    const float bn0 = bias[n0 + am];
    const float bn1 = bias[n0 + 16 + am];
    const int   mofs = (lane >> 4) << 3;   // 0 or 8

    #pragma unroll
    for (int r = 0; r < 8; ++r) {
        const size_t ma = (size_t)(m0 + mrow + r + mofs);        // subtile 0
        const size_t mbm = ma + 16;                              // subtile 1
        out[ma  * N_DIM + n0 + am]      = acc00[r] + bn0;
        out[ma  * N_DIM + n0 + 16 + am] = acc01[r] + bn1;
        out[mbm * N_DIM + n0 + am]      = acc10[r] + bn0;
        out[mbm * N_DIM + n0 + 16 + am] = acc11[r] + bn1;
    }
}

extern "C" void kernel_launch(void* const* d_in, const int* in_sizes, int n_in,
                              void* d_out, int out_size, void* d_ws, size_t ws_size,
                              hipStream_t stream) {
    const float* x    = (const float*)d_in[0];   // [2, 2048, 4096] f32
    const float* w    = (const float*)d_in[1];   // [4096, 4096] f32
    const float* bias = (const float*)d_in[2];   // [4096] f32
    const int*   mask = (const int*)d_in[3];     // [4096, 4096] block-constant

    float* out = (float*)d_out;                  // [2, 2048, 4096] f32

    dim3 grid(N_DIM / N_TILE, M_DIM / M_TILE);   // (128, 16)
    dim3 block(256);                             // 8 wave32
    bsl_wmma_f32_kernel<<<grid, block, 0, stream>>>(x, w, bias, mask, out);
}